// PairEmbedder_23081154248870
// MI455X (gfx1250) — compile-verified
//
#include <hip/hip_runtime.h>

typedef __attribute__((ext_vector_type(2))) float v2f;
typedef __attribute__((ext_vector_type(4))) float v4f;
typedef __attribute__((ext_vector_type(8))) float v8f;

#define H_DIM 256
#define W_DIM 512
#define CIN   23
#define COUT  64
#define B_DIM 4
#define HS    8              // H split factor (blocks along H)
#define HH    (H_DIM / HS)   // h values per wave

// ---------------- Kernel 0: zero the avg accumulator ----------------
__global__ __launch_bounds__(256) void zero_avg(float* __restrict__ avg, int n) {
    int i = blockIdx.x * 256 + threadIdx.x;
    if (i < n) avg[i] = 0.0f;
}

// ---------------- Kernel 1: WMMA conv + ReLU + mean over H -----------
// One wave (32 threads) per block. Block (wt, hz, b):
//   computes emb[64, w0..w0+15] for h in [hz*HH, hz*HH+HH), relu, sum,
//   atomically adds (sum/H) into avg[b][w][c]  (avg layout: (B, W, C), C fastest)
__global__ __launch_bounds__(32) void pair_avg_wmma(
    const float* __restrict__ msa,   // (B, CIN, H, W)
    const float* __restrict__ Wc,    // (COUT, CIN) row-major
    const float* __restrict__ bc,    // (COUT)
    float* __restrict__ avg)         // (B, W, COUT)
{
    const int lane = threadIdx.x;    // 0..31 (wave32)
    const int n    = lane & 15;      // N column (w) for B/C/D, M row for A
    const int g    = lane >> 4;      // half-wave select
    const int w0   = blockIdx.x * 16;
    const int hz   = blockIdx.y;
    const int b    = blockIdx.z;

    // --- Preload A tiles: a[t][j] = W_conv rows 16t..16t+15, K cols 4j..4j+3.
    // A 16x4 f32 layout: lanes 0-15 hold (M=lane, K={4j,4j+1}); lanes 16-31 (M=lane-16, K={4j+2,4j+3}).
    v2f a[4][6];
    #pragma unroll
    for (int t = 0; t < 4; ++t) {
        const int m = 16 * t + n;
        #pragma unroll
        for (int j = 0; j < 6; ++j) {
            const int k0 = 4 * j + 2 * g;
            a[t][j].x = (k0     < CIN) ? Wc[m * CIN + k0]     : 0.0f;
            a[t][j].y = (k0 + 1 < CIN) ? Wc[m * CIN + k0 + 1] : 0.0f;
        }
    }

    // --- Bias in C/D layout: tile t, reg r holds c_out = 16t + 8g + r.
    v8f bias[4];
    #pragma unroll
    for (int t = 0; t < 4; ++t)
        #pragma unroll
        for (int r = 0; r < 8; ++r)
            bias[t][r] = bc[16 * t + 8 * g + r];

    v8f sum[4];
    #pragma unroll
    for (int t = 0; t < 4; ++t)
        #pragma unroll
        for (int r = 0; r < 8; ++r)
            sum[t][r] = 0.0f;

    // msa element index: ((b*CIN + k)*H + h)*W + w
    const long base_bw = (long)b * CIN * H_DIM * W_DIM + (w0 + n);
    const int  h0 = hz * HH;

    for (int h = h0; h < h0 + HH; ++h) {
        // B 4x16 tile per K-step: reg.x = row K=4j+2g, reg.y = row K=4j+2g+1, col = n.
        v2f bm[6];
        #pragma unroll
        for (int j = 0; j < 6; ++j) {
            const int  k0 = 4 * j + 2 * g;            // max 22, always valid
            const long o0 = base_bw + ((long)k0 * H_DIM + h) * W_DIM;
            bm[j].x = msa[o0];
            bm[j].y = (k0 + 1 < CIN) ? msa[o0 + (long)H_DIM * W_DIM] : 0.0f;
        }
        #pragma unroll
        for (int t = 0; t < 4; ++t) {
            v8f c = bias[t];                           // C starts at the bias
            #pragma unroll
            for (int j = 0; j < 6; ++j)
                c = __builtin_amdgcn_wmma_f32_16x16x4_f32(
                        false, a[t][j], false, bm[j], (short)0, c, false, false);
            #pragma unroll
            for (int r = 0; r < 8; ++r) {
                // ReLU as select -> folds to a single v_max_num_f32
                // (maxnum(NaN,0)=0 and maxnum(-0,+0)=+0 match the select exactly)
                float rel = (c[r] > 0.0f) ? c[r] : 0.0f;
                sum[t][r] += rel;
            }
        }
    }

    const float scale = 1.0f / (float)H_DIM;           // mean over H
    float* dst = avg + (long)(b * W_DIM + w0 + n) * COUT;
    #pragma unroll
    for (int t = 0; t < 4; ++t)
        #pragma unroll
        for (int r = 0; r < 8; ++r)
            unsafeAtomicAdd(&dst[16 * t + 8 * g + r], sum[t][r] * scale);
}

// ---------------- Kernel 2: pair expansion (bandwidth-bound) ---------
// out[b][i][j][c] = avg[b][i][c] + avg[b][j][c], one float4 per thread.
// avg (512 KB) stays cache-resident; 256 MB output stored non-temporally
// (write-once stream larger than the 192 MB L2).
__global__ __launch_bounds__(256) void pair_expand(
    const float* __restrict__ avg,   // (B, W, COUT)
    float* __restrict__ out)         // (B, W, W, COUT)
{
    const long flat = (long)blockIdx.x * 256 + threadIdx.x;   // float4 index
    const int c4 = (int)(flat & 15);                          // COUT/4 = 16
    const int j  = (int)((flat >> 4)  & (W_DIM - 1));
    const int i  = (int)((flat >> 13) & (W_DIM - 1));
    const int b  = (int)(flat >> 22);

    const v4f* avg4 = (const v4f*)avg;
    v4f vi = avg4[((long)(b * W_DIM + i) << 4) + c4];
    v4f vj = avg4[((long)(b * W_DIM + j) << 4) + c4];
    v4f o  = vi + vj;
    __builtin_nontemporal_store(o, ((v4f*)out) + flat);
}

extern "C" void kernel_launch(void* const* d_in, const int* in_sizes, int n_in,
                              void* d_out, int out_size, void* d_ws, size_t ws_size,
                              hipStream_t stream) {
    const float* msa = (const float*)d_in[0];   // (4,23,256,512)
    const float* Wc  = (const float*)d_in[1];   // (64,23)
    const float* bc  = (const float*)d_in[2];   // (64,)
    float* out = (float*)d_out;                 // (4,512,512,64)
    float* avg = (float*)d_ws;                  // (4,512,64) = 512 KB scratch

    const int navg = B_DIM * W_DIM * COUT;
    zero_avg<<<(navg + 255) / 256, 256, 0, stream>>>(avg, navg);

    dim3 g1(W_DIM / 16, HS, B_DIM);             // 32 x 8 x 4 = 1024 waves
    pair_avg_wmma<<<g1, 32, 0, stream>>>(msa, Wc, bc, avg);

    const long n4 = (long)B_DIM * W_DIM * W_DIM * (COUT / 4);
    pair_expand<<<(unsigned)(n4 / 256), 256, 0, stream>>>(avg, out);
}